// PerformerAttention_28552942583828
// MI455X (gfx1250) — compile-verified
//
#include <hip/hip_runtime.h>
#include <math.h>

#define L_SEQ 2048
#define DMODEL 512
#define NHEADS 8
#define DHEAD 64
#define MFEAT 128
#define NCHUNKS 16
#define CHUNK 128
#define CHUNK_STRIDE 8320           // 128*64 state + 128 z
#define KERNEL_EPS 1e-4f
#define DEN_EPS 1e-6f
#define QK_SCALE 0.21022410381342864f   // 512^-0.25
#define NC_FEAT 0.08838834764831845f    // 128^-0.5

typedef float v2f __attribute__((ext_vector_type(2)));
typedef float v8f __attribute__((ext_vector_type(8)));

// ---------------------------------------------------------------------------
// Wave32 32x32 fp32 output block = 2x2 tiles of V_WMMA_F32_16X16X4_F32.
// Register blocking: per K-step load 2 A-frags + 2 B-frags, issue 4 WMMAs
// (2 floats loaded per lane per WMMA instead of 4 -> 2x arithmetic intensity,
// and 4 independent accumulator chains for the scheduler).
// A: element (i,k) at A[i*lda + k]      (caller offsets to block row origin)
// B: element (k,n) at B[k*bks + n*bns]  (caller offsets to block col origin)
// ISA layout (7.12.2): lanes 0-15 hold K pair {0,1}, lanes 16-31 pair {2,3};
// D vgpr i -> row i (lo lanes) / i+8 (hi lanes).
// ---------------------------------------------------------------------------
struct Acc2x2 { v8f c[2][2]; };

__device__ __forceinline__ Acc2x2 wmma_block2x2_f32(const float* __restrict__ A, int lda,
                                                    const float* __restrict__ B, int bks, int bns,
                                                    int K, int lane) {
  const int half = lane >> 4;
  const int l16  = lane & 15;
  const float* arow0 = A + l16 * lda;
  const float* arow1 = A + (l16 + 16) * lda;
  const float* bcol0 = B + l16 * bns;
  const float* bcol1 = B + (l16 + 16) * bns;
  const v8f zero = {0.f, 0.f, 0.f, 0.f, 0.f, 0.f, 0.f, 0.f};
  Acc2x2 r;
  r.c[0][0] = zero; r.c[0][1] = zero; r.c[1][0] = zero; r.c[1][1] = zero;
  for (int k0 = 0; k0 < K; k0 += 4) {
    const int ka = k0 + 2 * half;
    v2f a0, a1, b0, b1;
    a0.x = arow0[ka];          a0.y = arow0[ka + 1];
    a1.x = arow1[ka];          a1.y = arow1[ka + 1];
    b0.x = bcol0[ka * bks];    b0.y = bcol0[(ka + 1) * bks];
    b1.x = bcol1[ka * bks];    b1.y = bcol1[(ka + 1) * bks];
    r.c[0][0] = __builtin_amdgcn_wmma_f32_16x16x4_f32(false, a0, false, b0, (short)0, r.c[0][0], false, false);
    r.c[0][1] = __builtin_amdgcn_wmma_f32_16x16x4_f32(false, a0, false, b1, (short)0, r.c[0][1], false, false);
    r.c[1][0] = __builtin_amdgcn_wmma_f32_16x16x4_f32(false, a1, false, b0, (short)0, r.c[1][0], false, false);
    r.c[1][1] = __builtin_amdgcn_wmma_f32_16x16x4_f32(false, a1, false, b1, (short)0, r.c[1][1], false, false);
  }
  return r;
}

// ---- Stage 1: Q/K/V projections (y = x @ W^T + b), QK pre-scaled ----------
__global__ void __launch_bounds__(32)
perf_qkv_proj(const float* __restrict__ xq, const float* __restrict__ xk,
              const float* __restrict__ xv,
              const float* __restrict__ Wq, const float* __restrict__ bq,
              const float* __restrict__ Wk, const float* __restrict__ bk,
              const float* __restrict__ Wv, const float* __restrict__ bv,
              float* __restrict__ Q, float* __restrict__ K, float* __restrict__ V) {
  const int tm = blockIdx.x * 32;
  const int tn = blockIdx.y * 32;
  const int mat = blockIdx.z;
  const int lane = threadIdx.x;
  const float* X; const float* W; const float* bias; float* Y; float scl;
  if (mat == 0)      { X = xq; W = Wq; bias = bq; Y = Q; scl = QK_SCALE; }
  else if (mat == 1) { X = xk; W = Wk; bias = bk; Y = K; scl = QK_SCALE; }
  else               { X = xv; W = Wv; bias = bv; Y = V; scl = 1.f; }
  // B[k][n] = W[n][k]  ->  bks = 1, bns = DMODEL
  Acc2x2 r = wmma_block2x2_f32(X + tm * DMODEL, DMODEL,
                               W + tn * DMODEL, 1, DMODEL, DMODEL, lane);
  const int l16 = lane & 15;
  const int half = lane >> 4;
#pragma unroll
  for (int ti = 0; ti < 2; ++ti)
#pragma unroll
    for (int tj = 0; tj < 2; ++tj) {
      const int col = tn + tj * 16 + l16;
      const int rbase = tm + ti * 16 + 8 * half;
      const float bb = bias[col];
#pragma unroll
      for (int i = 0; i < 8; ++i)
        Y[(rbase + i) * DMODEL + col] = (r.c[ti][tj][i] + bb) * scl;
    }
}

// ---- Stage 2a: h_k[h,l] = -0.5 * ||k_hl||^2 -------------------------------
__global__ void __launch_bounds__(256)
perf_hk(const float* __restrict__ K, float* __restrict__ hk) {
  const int idx = blockIdx.x * 256 + threadIdx.x;   // = h*2048 + l
  const int h = idx >> 11;
  const int l = idx & (L_SEQ - 1);
  const float* row = K + l * DMODEL + h * DHEAD;
  float s = 0.f;
#pragma unroll 8
  for (int d = 0; d < DHEAD; ++d) s += row[d] * row[d];
  hk[idx] = -0.5f * s;
}

// ---- Stage 2b: k_stab = max over all h_k (single block reduce) ------------
__global__ void __launch_bounds__(256)
perf_kstab(const float* __restrict__ hk, float* __restrict__ kstab) {
  __shared__ float red[256];
  const int t = threadIdx.x;
  float m = -3.402823e38f;
  for (int i = t; i < NHEADS * L_SEQ; i += 256) m = fmaxf(m, hk[i]);
  red[t] = m;
  __syncthreads();
  for (int s = 128; s > 0; s >>= 1) {
    if (t < s) red[t] = fmaxf(red[t], red[t + s]);
    __syncthreads();
  }
  if (t == 0) kstab[0] = red[0];
}

// ---- Stage 2c: feature projections + exp kernel --------------------------
// q' = nc*(exp(proj_q) + eps)                 (h_q stabilizer cancels)
// k' = nc*(exp(h_k + proj_k - k_stab) + eps)
__global__ void __launch_bounds__(32)
perf_primes(const float* __restrict__ Q, const float* __restrict__ K,
            const float* __restrict__ RF, const float* __restrict__ hk,
            const float* __restrict__ kstab,
            float* __restrict__ qprime, float* __restrict__ kprime) {
  const int tm = blockIdx.x * 32;            // l block
  const int tn = blockIdx.y * 32;            // m block
  const int h = blockIdx.z >> 1;
  const int isK = blockIdx.z & 1;
  const int lane = threadIdx.x;
  const float* A = (isK ? K : Q) + tm * DMODEL + h * DHEAD;
  // RF row-major (64 x 128): B[k][n] at RF[k*128 + n]
  Acc2x2 r = wmma_block2x2_f32(A, DMODEL, RF + tn, MFEAT, 1, DHEAD, lane);
  const int l16 = lane & 15;
  const int half = lane >> 4;
  const float ks = isK ? kstab[0] : 0.f;
#pragma unroll
  for (int ti = 0; ti < 2; ++ti)
#pragma unroll
    for (int tj = 0; tj < 2; ++tj) {
      const int mcol = tn + tj * 16 + l16;
      const int rbase = tm + ti * 16 + 8 * half;
      if (isK) {
#pragma unroll
        for (int i = 0; i < 8; ++i) {
          const int l = rbase + i;
          kprime[(h * L_SEQ + l) * MFEAT + mcol] =
              NC_FEAT * (expf(hk[h * L_SEQ + l] + r.c[ti][tj][i] - ks) + KERNEL_EPS);
        }
      } else {
#pragma unroll
        for (int i = 0; i < 8; ++i) {
          const int l = rbase + i;
          qprime[(h * L_SEQ + l) * MFEAT + mcol] =
              NC_FEAT * (expf(r.c[ti][tj][i]) + KERNEL_EPS);
        }
      }
    }
}

// ---- Stage 3a: per-(head,chunk) sums of k'(x)v and k' --------------------
__global__ void __launch_bounds__(256)
perf_chunk_sums(const float* __restrict__ kprime, const float* __restrict__ V,
                float* __restrict__ chunkbuf) {
  const int h = blockIdx.x >> 4;
  const int c = blockIdx.x & 15;
  const int t = threadIdx.x;
  __shared__ float kp[MFEAT];
  __shared__ float vv[DHEAD];
  const int m = t >> 1;
  const int d0 = (t & 1) * 32;
  float acc[32];
#pragma unroll
  for (int j = 0; j < 32; ++j) acc[j] = 0.f;
  float zacc = 0.f;
  for (int l0 = 0; l0 < CHUNK; ++l0) {
    const int l = c * CHUNK + l0;
    if (t < MFEAT) kp[t] = kprime[(h * L_SEQ + l) * MFEAT + t];
    if (t < DHEAD) vv[t] = V[l * DMODEL + h * DHEAD + t];
    __syncthreads();
    const float kpm = kp[m];
#pragma unroll
    for (int j = 0; j < 32; ++j) acc[j] += kpm * vv[d0 + j];
    zacc += kpm;
    __syncthreads();
  }
  float* buf = chunkbuf + (size_t)(h * NCHUNKS + c) * CHUNK_STRIDE;
#pragma unroll
  for (int j = 0; j < 32; ++j) buf[m * DHEAD + d0 + j] = acc[j];
  if ((t & 1) == 0) buf[MFEAT * DHEAD + m] = zacc;
}

// ---- Stage 3b: exclusive prefix of chunk sums across chunks (per head) ---
__global__ void __launch_bounds__(256)
perf_chunk_prefix(float* __restrict__ chunkbuf) {
  const int h = blockIdx.x;
  for (int idx = threadIdx.x; idx < CHUNK_STRIDE; idx += 256) {
    float carry = 0.f;
    for (int c = 0; c < NCHUNKS; ++c) {
      float* p = chunkbuf + (size_t)(h * NCHUNKS + c) * CHUNK_STRIDE + idx;
      const float tmp = *p;
      *p = carry;
      carry += tmp;
    }
  }
}

// ---- Stage 3c: in-chunk sequential scan; 32KB state in LDS ---------------
__global__ void __launch_bounds__(256)
perf_scan(const float* __restrict__ qprime, const float* __restrict__ kprime,
          const float* __restrict__ V, const float* __restrict__ chunkbuf,
          float* __restrict__ context) {
  const int h = blockIdx.x >> 4;
  const int c = blockIdx.x & 15;
  const int t = threadIdx.x;
  __shared__ float S[MFEAT * DHEAD];
  __shared__ float z[MFEAT];
  __shared__ float qp[MFEAT];
  __shared__ float kp[MFEAT];
  __shared__ float vv[DHEAD];
  __shared__ float red[256];
  __shared__ float dred[MFEAT];
  const float* buf = chunkbuf + (size_t)(h * NCHUNKS + c) * CHUNK_STRIDE;
  for (int i = t; i < MFEAT * DHEAD; i += 256) S[i] = buf[i];
  if (t < MFEAT) z[t] = buf[MFEAT * DHEAD + t];
  __syncthreads();
  const int m = t >> 1;
  const int d0 = (t & 1) * 32;
  const int dd = t & 63;
  const int part = t >> 6;
  for (int l0 = 0; l0 < CHUNK; ++l0) {
    const int l = c * CHUNK + l0;
    if (t < MFEAT) qp[t] = qprime[(h * L_SEQ + l) * MFEAT + t];
    else           kp[t - MFEAT] = kprime[(h * L_SEQ + l) * MFEAT + (t - MFEAT)];
    if (t < DHEAD) vv[t] = V[l * DMODEL + h * DHEAD + t];
    __syncthreads();
    // inclusive update: S += k' (x) v ; z += k'
    const float kpm = kp[m];
#pragma unroll
    for (int j = 0; j < 32; ++j) S[m * DHEAD + d0 + j] += kpm * vv[d0 + j];
    if ((t & 1) == 0) z[m] += kpm;
    __syncthreads();
    // numerator partials: 4 partial sums over m per output d
    float s = 0.f;
    const int mb = part * 32;
#pragma unroll
    for (int j = 0; j < 32; ++j) s += qp[mb + j] * S[(mb + j) * DHEAD + dd];
    red[t] = s;
    if (t < MFEAT) dred[t] = qp[t] * z[t];
    __syncthreads();
    if (t < 64) dred[t] += dred[t + 64];
    __syncthreads();
    if (t == 0) {
      float s2 = 0.f;
      for (int j = 0; j < 64; ++j) s2 += dred[j];
      dred[0] = s2;
    }
    __syncthreads();
    if (t < DHEAD) {
      const float num = red[dd] + red[64 + dd] + red[128 + dd] + red[192 + dd];
      context[l * DMODEL + h * DHEAD + dd] = num / (dred[0] + DEN_EPS);
    }
    __syncthreads();
  }
}

// ---- Stage 4: output projection ------------------------------------------
__global__ void __launch_bounds__(32)
perf_out_proj(const float* __restrict__ ctx, const float* __restrict__ Wo,
              const float* __restrict__ bo, float* __restrict__ out) {
  const int tm = blockIdx.x * 32;
  const int tn = blockIdx.y * 32;
  const int lane = threadIdx.x;
  Acc2x2 r = wmma_block2x2_f32(ctx + tm * DMODEL, DMODEL,
                               Wo + tn * DMODEL, 1, DMODEL, DMODEL, lane);
  const int l16 = lane & 15;
  const int half = lane >> 4;
#pragma unroll
  for (int ti = 0; ti < 2; ++ti)
#pragma unroll
    for (int tj = 0; tj < 2; ++tj) {
      const int col = tn + tj * 16 + l16;
      const int rbase = tm + ti * 16 + 8 * half;
      const float bb = bo[col];
#pragma unroll
      for (int i = 0; i < 8; ++i)
        out[(rbase + i) * DMODEL + col] = r.c[ti][tj][i] + bb;
    }
}

extern "C" void kernel_launch(void* const* d_in, const int* in_sizes, int n_in,
                              void* d_out, int out_size, void* d_ws, size_t ws_size,
                              hipStream_t stream) {
  (void)in_sizes; (void)n_in; (void)out_size; (void)ws_size;
  const float* query = (const float*)d_in[0];
  const float* key_  = (const float*)d_in[1];
  const float* value = (const float*)d_in[2];
  const float* Wq = (const float*)d_in[3];
  const float* bq = (const float*)d_in[4];
  const float* Wk = (const float*)d_in[5];
  const float* bk = (const float*)d_in[6];
  const float* Wv = (const float*)d_in[7];
  const float* bv = (const float*)d_in[8];
  const float* Wo = (const float*)d_in[9];
  const float* bo = (const float*)d_in[10];
  const float* RF = (const float*)d_in[11];

  float* ws = (float*)d_ws;
  const size_t SZ_LD = (size_t)L_SEQ * DMODEL;                 // 1M floats
  const size_t SZ_PRIME = (size_t)NHEADS * L_SEQ * MFEAT;      // 2M floats
  float* Q      = ws;
  float* K      = Q + SZ_LD;
  float* V      = K + SZ_LD;
  float* hk     = V + SZ_LD;                                   // 16384
  float* kstab  = hk + (size_t)NHEADS * L_SEQ;                 // 1 (padded 64)
  float* qprime = kstab + 64;
  float* kprime = qprime + SZ_PRIME;
  float* chunkbuf = kprime + SZ_PRIME;  // 8*16*8320 floats
  float* context = Q;                   // alias: Q dead after perf_primes

  perf_qkv_proj<<<dim3(L_SEQ / 32, DMODEL / 32, 3), 32, 0, stream>>>(
      query, key_, value, Wq, bq, Wk, bk, Wv, bv, Q, K, V);
  perf_hk<<<(NHEADS * L_SEQ) / 256, 256, 0, stream>>>(K, hk);
  perf_kstab<<<1, 256, 0, stream>>>(hk, kstab);
  perf_primes<<<dim3(L_SEQ / 32, MFEAT / 32, 2 * NHEADS), 32, 0, stream>>>(
      Q, K, RF, hk, kstab, qprime, kprime);
  perf_chunk_sums<<<NHEADS * NCHUNKS, 256, 0, stream>>>(kprime, V, chunkbuf);
  perf_chunk_prefix<<<NHEADS, 256, 0, stream>>>(chunkbuf);
  perf_scan<<<NHEADS * NCHUNKS, 256, 0, stream>>>(qprime, kprime, V, chunkbuf,
                                                  context);
  perf_out_proj<<<dim3(L_SEQ / 32, DMODEL / 32), 32, 0, stream>>>(
      context, Wo, bo, (float*)d_out);
}